// MeshSDFLoss_84705345012280
// MI455X (gfx1250) — compile-verified
//
#include <hip/hip_runtime.h>

typedef __attribute__((ext_vector_type(2))) float v2f;
typedef __attribute__((ext_vector_type(8))) float v8f;

// Tile-packed staging block: one 16-triangle tile = 104 float4 (1664 B):
//   [0..31]  : B-operands per (tri, K-half): (Bab.x,Bab.y,Bac.x,Bac.y)
//              half0 = (abx,aby,acx,acy), half1 = (abz,0,acz,0)
//   [32..47] : tc0 = (ab.a, ac.a, ab.b, ac.b)
//   [48..63] : tc1 = (ab.c, ac.c, |a|^2, |b|^2)
//   [64..79] : tc2 = (|c|^2, |ab|^2, |ac|^2, ab.ac)
//   [80..95] : tc3 = (rden_ab, rden_ac, rden_bc, |bc|^2)
//   [96..111]: A-vertex B-operand per (tri, K-half) as float2 (overruns TILE_F4):
//              half0 = (ax,ay), half1 = (az,0)
#define TILE_F4 104

// ---------------------------------------------------------------------------
// Kernel 1: per-triangle precompute into the tile-packed staging stream.
// ---------------------------------------------------------------------------
__global__ void tri_precompute(const float* __restrict__ verts,
                               const int*   __restrict__ faces,
                               float4* __restrict__ stage,
                               int F)
{
    int f = blockIdx.x * blockDim.x + threadIdx.x;
    if (f >= F) return;
    int ia = faces[3*f+0], ib = faces[3*f+1], ic = faces[3*f+2];
    float ax = verts[3*ia+0], ay = verts[3*ia+1], az = verts[3*ia+2];
    float bx = verts[3*ib+0], by = verts[3*ib+1], bz = verts[3*ib+2];
    float cx = verts[3*ic+0], cy = verts[3*ic+1], cz = verts[3*ic+2];
    float abx = bx-ax, aby = by-ay, abz = bz-az;
    float acx = cx-ax, acy = cy-ay, acz = cz-az;
    float bcx = cx-bx, bcy = cy-by, bcz = cz-bz;
    float ab_a = abx*ax + aby*ay + abz*az;
    float ac_a = acx*ax + acy*ay + acz*az;
    float ab_b = abx*bx + aby*by + abz*bz;
    float ac_b = acx*bx + acy*by + acz*bz;
    float ab_c = abx*cx + aby*cy + abz*cz;
    float ac_c = acx*cx + acy*cy + acz*cz;
    float aa = ax*ax+ay*ay+az*az;
    float bb = bx*bx+by*by+bz*bz;
    float cc = cx*cx+cy*cy+cz*cz;
    float abab = abx*abx+aby*aby+abz*abz;
    float acac = acx*acx+acy*acy+acz*acz;
    float abac = abx*acx+aby*acy+abz*acz;
    float bcbc = bcx*bcx+bcy*bcy+bcz*bcz;
    // Reference denominators are per-triangle constants:
    //   d1-d3 = |ab|^2 ; d2-d6 = |ac|^2 ; (d4-d3)+(d5-d6) = |bc|^2
    // with the reference's (den==0 -> divide by 1) guard folded in.
    float rab = (abab != 0.f) ? 1.f/abab : 1.f;
    float rac = (acac != 0.f) ? 1.f/acac : 1.f;
    float rbc = (bcbc != 0.f) ? 1.f/bcbc : 1.f;

    const int tile = f >> 4;
    const int j    = f & 15;
    float4* base = stage + (size_t)tile * TILE_F4;
    base[2*j+0] = make_float4(abx, aby, acx, acy);   // K-half 0
    base[2*j+1] = make_float4(abz, 0.f, acz, 0.f);   // K-half 1
    base[32+j]  = make_float4(ab_a, ac_a, ab_b, ac_b);
    base[48+j]  = make_float4(ab_c, ac_c, aa,   bb);
    base[64+j]  = make_float4(cc,   abab, acac, abac);
    base[80+j]  = make_float4(rab,  rac,  rbc,  bcbc);
    float2* ba = (float2*)(base + 96);
    ba[2*j+0] = make_float2(ax, ay);                  // K-half 0
    ba[2*j+1] = make_float2(az, 0.f);                 // K-half 1
}

// ---------------------------------------------------------------------------
// Kernel 2: WMMA point-triangle sqdist with double-buffered async LDS staging.
// 128 threads = 4 waves = 4 point-tiles per block; all waves consume the same
// triangle-tile stream from LDS (loaded once per block via ASYNC loads).
// ---------------------------------------------------------------------------
__global__ __launch_bounds__(128) void sdf_wmma(
    const float*  __restrict__ points,
    const float4* __restrict__ stage,
    float* __restrict__ dist_out, int* __restrict__ assoc_out,
    int nPtTiles, int nTriTiles)
{
    __shared__ alignas(16) float4 sbuf[2][TILE_F4];

    const int tid  = threadIdx.x;
    const int lane = tid & 31;
    const int wv   = tid >> 5;
    int tile = blockIdx.x * 4 + wv;
    const bool valid = (tile < nPtTiles);
    if (!valid) tile = nPtTiles - 1;     // keep all waves in the barrier dance
    const int m    = lane & 15;          // triangle column / point id in tile
    const int half = lane >> 4;          // K-half for A/B operands
    const int ptBase = tile * 16;

    // Issue async staging loads for triangle tile t into LDS buffer p.
    auto issue_tile = [&](int t, int p) {
        if (tid < TILE_F4) {
            uint32_t lds = (uint32_t)(uintptr_t)&sbuf[p][tid];
            const char* gb = (const char*)(stage + (size_t)t * TILE_F4);
            uint32_t go = (uint32_t)tid * 16u;
            asm volatile("global_load_async_to_lds_b128 %0, %1, %2"
                         :: "v"(lds), "v"(go), "s"(gb) : "memory");
        }
    };

    issue_tile(0, 0);   // prologue: stage tile 0 (overlaps with A setup below)

    // A operand: lanes 0-15 hold (x,y)=K0,K1 of point m; lanes 16-31 hold
    // (z,0)=K2,K3 of the same point (ISA 16x4 f32 A layout).
    const float px = points[3*(ptBase+m)+0];
    const float py = points[3*(ptBase+m)+1];
    const float pz = points[3*(ptBase+m)+2];
    const float pp = px*px + py*py + pz*pz;
    v2f Aop;
    Aop.x = half ? pz  : px;
    Aop.y = half ? 0.f : py;

    // |p|^2 for the 8 point-rows this lane owns in the D layout (M = r+8*half)
    float ppr[8];
    #pragma unroll
    for (int r = 0; r < 8; ++r)
        ppr[r] = __shfl(pp, r + 8*half, 32);

    float bd[8];
    int   bi[8];
    #pragma unroll
    for (int r = 0; r < 8; ++r) { bd[r] = 3.402823e38f; bi[r] = 0; }

    asm volatile("s_wait_asynccnt 0x0" ::: "memory");
    __syncthreads();

    for (int t = 0; t < nTriTiles; ++t) {
        const int p = t & 1;
        if (t + 1 < nTriTiles) issue_tile(t + 1, p ^ 1);   // prefetch next tile

        const float4* tl = &sbuf[p][0];

        // Per-lane B operands straight from LDS (half-selection is an address)
        const float4 bp4 = tl[2*m + half];
        v2f Bab; Bab.x = bp4.x; Bab.y = bp4.y;
        v2f Bac; Bac.x = bp4.z; Bac.y = bp4.w;
        const float2 ba2 = ((const float2*)(tl + 96))[2*m + half];
        v2f Ba;  Ba.x = ba2.x;  Ba.y = ba2.y;

        const v8f zacc = {0.f,0.f,0.f,0.f,0.f,0.f,0.f,0.f};
        // s1 = ab.p, s2 = ac.p, s3 = a.p over the 16x16 (point,tri) tile
        v8f s1 = __builtin_amdgcn_wmma_f32_16x16x4_f32(
                     false, Aop, false, Bab, (short)0, zacc, false, false);
        v8f s2 = __builtin_amdgcn_wmma_f32_16x16x4_f32(
                     false, Aop, false, Bac, (short)0, zacc, false, false);
        v8f s3 = __builtin_amdgcn_wmma_f32_16x16x4_f32(
                     false, Aop, false, Ba,  (short)0, zacc, false, false);

        const float4 c0 = tl[32+m];
        const float4 c1 = tl[48+m];
        const float4 c2 = tl[64+m];
        const float4 c3 = tl[80+m];
        const int tr = t*16 + m;

        #pragma unroll
        for (int r = 0; r < 8; ++r) {
            const float S1 = s1[r], S2 = s2[r], S3 = s3[r], PP = ppr[r];
            const float d1 = S1 - c0.x;
            const float d2 = S2 - c0.y;
            const float d3 = S1 - c0.z;
            const float d4 = S2 - c0.w;
            const float d5 = S1 - c1.x;
            const float d6 = S2 - c1.y;
            const float ap2 = fmaf(-2.f, S3, PP) + c1.z;           // |p-a|^2
            const float bp2 = fmaf(-2.f, S3 + S1, PP) + c1.w;      // |p-b|^2
            const float cp2 = fmaf(-2.f, S3 + S2, PP) + c2.x;      // |p-c|^2
            const float vc_ = fmaf(d1, d4, -d3*d2);
            const float vb_ = fmaf(d5, d2, -d1*d6);
            const float va_ = fmaf(d3, d6, -d5*d4);
            const float den = va_ + vb_ + vc_;
            const float dsel = (den != 0.f) ? den : 1.f;
            // fast reciprocal + 1 Newton step instead of IEEE divide
            float inv = __builtin_amdgcn_rcpf(dsel);
            inv = inv * fmaf(-dsel, inv, 2.f);
            const float v = vb_ * inv;
            const float w = vc_ * inv;
            // interior: |ap - v*ab - w*ac|^2
            float dist = ap2 - 2.f*fmaf(v, d1, w*d2)
                       + fmaf(v, fmaf(v, c2.y, 2.f*w*c2.w), w*w*c2.z);
            const float t43 = d4 - d3;
            const float t56 = d5 - d6;
            const float wbc  = t43 * c3.z;
            const float dbc  = fmaf(wbc,  fmaf(wbc,  c3.w, -2.f*t43), bp2);
            const float wac  = d2  * c3.y;
            const float dac  = fmaf(wac,  fmaf(wac,  c2.z, -2.f*d2),  ap2);
            const float vab_ = d1  * c3.x;
            const float dab  = fmaf(vab_, fmaf(vab_, c2.y, -2.f*d1),  ap2);
            // region overrides in the reference's order (later wins)
            if (va_ <= 0.f && t43 >= 0.f && t56 >= 0.f) dist = dbc;
            if (vb_ <= 0.f && d2 >= 0.f && d6 <= 0.f)   dist = dac;
            if (vc_ <= 0.f && d1 >= 0.f && d3 <= 0.f)   dist = dab;
            if (d6 >= 0.f && d5 <= d6)                  dist = cp2;
            if (d3 >= 0.f && d4 <= d3)                  dist = bp2;
            if (d1 <= 0.f && d2 <= 0.f)                 dist = ap2;

            if (dist < bd[r]) { bd[r] = dist; bi[r] = tr; }  // ties keep lower tr
        }

        asm volatile("s_wait_asynccnt 0x0" ::: "memory");  // t+1 staged
        __syncthreads();                                    // all readers done
    }

    // Min/argmin across the 16 triangle columns (lanes within each half).
    #pragma unroll
    for (int r = 0; r < 8; ++r) {
        float d = bd[r];
        int   i = bi[r];
        #pragma unroll
        for (int off = 1; off < 16; off <<= 1) {
            const float od = __shfl_xor(d, off, 32);
            const int   oi = __shfl_xor(i, off, 32);
            if (od < d || (od == d && oi < i)) { d = od; i = oi; }
        }
        if (m == 0 && valid) {
            const int pt = ptBase + r + 8*half;
            dist_out[pt]  = d;
            assoc_out[pt] = i;
        }
    }
}

// ---------------------------------------------------------------------------
// Kernel 3: deterministic fixed-order loss reduction (no float atomics).
// ---------------------------------------------------------------------------
__global__ void loss_reduce(const float* __restrict__ dist,
                            float* __restrict__ out, int P)
{
    __shared__ float sm[256];
    float s = 0.f;
    for (int i = threadIdx.x; i < P; i += 256) s += dist[i];
    sm[threadIdx.x] = s;
    __syncthreads();
    for (int k = 128; k > 0; k >>= 1) {
        if ((int)threadIdx.x < k) sm[threadIdx.x] += sm[threadIdx.x + k];
        __syncthreads();
    }
    if (threadIdx.x == 0) out[0] = sm[0];
}

// ---------------------------------------------------------------------------
extern "C" void kernel_launch(void* const* d_in, const int* in_sizes, int n_in,
                              void* d_out, int out_size, void* d_ws, size_t ws_size,
                              hipStream_t stream)
{
    const float* verts  = (const float*)d_in[0];
    const int*   faces  = (const int*)  d_in[1];
    const float* points = (const float*)d_in[2];
    const int F = in_sizes[1] / 3;
    const int P = in_sizes[2] / 3;

    float4* stage = (float4*)d_ws;   // (F/16) * 104 float4 = ~1.7 MB

    float* out       = (float*)d_out;
    float* dist_out  = out + 1;
    int*   assoc_out = ((int*)d_out) + 1 + P;

    tri_precompute<<<(F + 255)/256, 256, 0, stream>>>(verts, faces, stage, F);

    const int nPtTiles  = P / 16;
    const int nTriTiles = F / 16;
    sdf_wmma<<<(nPtTiles + 3)/4, 128, 0, stream>>>(
        points, stage, dist_out, assoc_out, nPtTiles, nTriTiles);

    loss_reduce<<<1, 256, 0, stream>>>(dist_out, out, P);
}